// QuantumLayer_49392123904178
// MI455X (gfx1250) — compile-verified
//
#include <hip/hip_runtime.h>
#include <hip/hip_bf16.h>

typedef __attribute__((ext_vector_type(2))) float v2f;
typedef __attribute__((ext_vector_type(8))) float v8f;

// ---------------------------------------------------------------------------
// Kernel 1: build the 64x64 real observable matrix A = Re(W^dagger Z0 W)
// from the 24 weights. W = product of the w-dependent gates.
// Qubit q <-> bit (5-q) of the flat 64-index (qubit 0 = MSB, stride 32).
// One block of 64 threads; thread j owns column j of W in LDS.
// ---------------------------------------------------------------------------
__global__ __launch_bounds__(64) void build_A_kernel(const float* __restrict__ w,
                                                     float* __restrict__ A) {
    __shared__ float Wr[64 * 64];
    __shared__ float Wi[64 * 64];
    const int j = threadIdx.x;

    for (int k = 0; k < 64; ++k) {
        Wr[k * 64 + j] = (k == j) ? 1.0f : 0.0f;
        Wi[k * 64 + j] = 0.0f;
    }

    auto RY = [&](float th, int q) {
        float s, c; __sincosf(0.5f * th, &s, &c);
        int st = 1 << (5 - q);
        for (int t = 0; t < 64; ++t) if (!(t & st)) {
            int u = t | st;
            float a0 = Wr[t*64+j], a1 = Wr[u*64+j];
            Wr[t*64+j] = c*a0 - s*a1;  Wr[u*64+j] = s*a0 + c*a1;
            float b0 = Wi[t*64+j], b1 = Wi[u*64+j];
            Wi[t*64+j] = c*b0 - s*b1;  Wi[u*64+j] = s*b0 + c*b1;
        }
    };
    auto RZ = [&](float th, int q) {
        float s, c; __sincosf(0.5f * th, &s, &c);
        int st = 1 << (5 - q);
        for (int t = 0; t < 64; ++t) {
            float ci = (t & st) ? s : -s;          // exp(+-i th/2)
            float ar = Wr[t*64+j], ai = Wi[t*64+j];
            Wr[t*64+j] = ar*c - ai*ci;
            Wi[t*64+j] = ar*ci + ai*c;
        }
    };
    auto CNOT = [&](int cq, int tq) {
        int sc = 1 << (5 - cq), st = 1 << (5 - tq);
        for (int t = 0; t < 64; ++t) if ((t & sc) && !(t & st)) {
            int u = t | st;
            float tmp = Wr[t*64+j]; Wr[t*64+j] = Wr[u*64+j]; Wr[u*64+j] = tmp;
            tmp = Wi[t*64+j]; Wi[t*64+j] = Wi[u*64+j]; Wi[u*64+j] = tmp;
        }
    };
    auto CRY = [&](float th, int cq, int tq) {
        float s, c; __sincosf(0.5f * th, &s, &c);
        int sc = 1 << (5 - cq), st = 1 << (5 - tq);
        for (int t = 0; t < 64; ++t) if ((t & sc) && !(t & st)) {
            int u = t | st;
            float a0 = Wr[t*64+j], a1 = Wr[u*64+j];
            Wr[t*64+j] = c*a0 - s*a1;  Wr[u*64+j] = s*a0 + c*a1;
            float b0 = Wi[t*64+j], b1 = Wi[u*64+j];
            Wi[t*64+j] = c*b0 - s*b1;  Wi[u*64+j] = s*b0 + c*b1;
        }
    };

    for (int i = 0; i < 6; ++i) RY(w[i], i);
    for (int i = 0; i < 6; ++i) CNOT(i, (i + 1) % 6);
    for (int q = 0; q < 3; ++q) CRY(w[6 + q], 2*q + 1, 2*q);
    const int act[3] = {0, 2, 4};
    for (int i = 0; i < 3; ++i) RY(w[9 + i], act[i]);
    for (int i = 0; i < 3; ++i) RZ(w[12 + i], act[i]);
    const int cts[3][2] = {{0, 2}, {2, 4}, {4, 0}};
    for (int i = 0; i < 3; ++i) CRY(w[15 + i], cts[i][0], cts[i][1]);
    for (int i = 0; i < 3; ++i) RY(w[18 + i], act[i]);
    RZ(w[21], 0);
    CRY(w[22], 2, 0);
    CRY(w[23], 4, 0);

    __syncthreads();

    // A[i][jj] = sum_k z_k * (Wr[k][i]*Wr[k][jj] + Wi[k][i]*Wi[k][jj]),
    // z_k = +1 if qubit0(k)==0 else -1 (qubit 0 = bit 5).
    const int i = threadIdx.x;
    for (int jj = 0; jj < 64; ++jj) {
        float acc = 0.0f;
        for (int k = 0; k < 64; ++k) {
            float z = (k & 32) ? -1.0f : 1.0f;
            acc += z * (Wr[k*64 + i] * Wr[k*64 + jj] + Wi[k*64 + i] * Wi[k*64 + jj]);
        }
        A[i * 64 + jj] = acc;
    }
}

// ---------------------------------------------------------------------------
// Kernel 2: per-patch real state psi (64 floats, data-dependent part), then
// ev = psi^T A psi as a WMMA GEMM Q = A * Psi (per-wave 64x32 tile, two
// 16-column groups), then per-column dot product.
// Grid 961 x 128 threads covers 128*31*31 = 123008 patches exactly.
// ---------------------------------------------------------------------------
__global__ __launch_bounds__(128) void qev_kernel(const float* __restrict__ inp,
                                                  const float* __restrict__ A,
                                                  float* __restrict__ out) {
    __shared__ float Psi[4 * 64 * 32];          // per-wave 64x32 tile
    const int lane = threadIdx.x & 31;
    const int wv   = threadIdx.x >> 5;
    const int p    = blockIdx.x * 128 + threadIdx.x;   // global patch id

    // ---- gather the 12 patch angles (x[c*4+d] = inputs[b, i+dh, j+dw, c]) --
    const int b  = p / 961;
    const int hw = p - b * 961;
    const int ii = hw / 31;
    const int jj = hw - ii * 31;
    const float* base = inp + (((b * 32 + ii) * 32) + jj) * 3;
    float x[12];
#pragma unroll
    for (int c = 0; c < 3; ++c) {
        x[c * 4 + 0] = base[c];                  // dh=0 dw=0
        x[c * 4 + 1] = base[3 + c];              // dh=0 dw=1
        x[c * 4 + 2] = base[96 + c];             // dh=1 dw=0  (32*3)
        x[c * 4 + 3] = base[99 + c];             // dh=1 dw=1
    }

    // ---- psi after first RY layer: product state (all real) ----------------
    float psi[64];
    {
        float cs[6], sn[6];
#pragma unroll
        for (int q = 0; q < 6; ++q) __sincosf(0.5f * x[q], &sn[q], &cs[q]);
#pragma unroll
        for (int s = 0; s < 64; ++s) {
            float v = 1.0f;
#pragma unroll
            for (int q = 0; q < 6; ++q) v *= ((s >> (5 - q)) & 1) ? sn[q] : cs[q];
            psi[s] = v;
        }
    }
    // ---- CNOT ring ---------------------------------------------------------
#pragma unroll
    for (int i = 0; i < 6; ++i) {
        const int cq = i, tq = (i + 1) % 6;
        const int sc = 1 << (5 - cq), st = 1 << (5 - tq);
#pragma unroll
        for (int s = 0; s < 64; ++s)
            if ((s & sc) && !(s & st)) {
                int u = s | st;
                float tmp = psi[s]; psi[s] = psi[u]; psi[u] = tmp;
            }
    }
    // ---- second RY layer ---------------------------------------------------
#pragma unroll
    for (int q = 0; q < 6; ++q) {
        float s2, c2; __sincosf(0.5f * x[6 + q], &s2, &c2);
        const int st = 1 << (5 - q);
#pragma unroll
        for (int s = 0; s < 64; ++s)
            if (!(s & st)) {
                int u = s | st;
                float a0 = psi[s], a1 = psi[u];
                psi[s] = c2 * a0 - s2 * a1;
                psi[u] = s2 * a0 + c2 * a1;
            }
    }

    // ---- stage Psi tile in LDS: Psi[k][lane] for this wave -----------------
    float* myPsi = &Psi[wv * 2048];
#pragma unroll
    for (int k = 0; k < 64; ++k) myPsi[k * 32 + lane] = psi[k];
    __syncthreads();

    // ---- Q = A * Psi via V_WMMA_F32_16X16X4_F32 ----------------------------
    // A frag: lane L -> row M = r*16 + (L&15), K = t*4 + 2*(L>=16) + v
    // B frag: lane L -> K = t*4 + 2*(L>=16) + v, col N = (L&15)
    // C/D   : lane L -> row M = v + 8*(L>=16), col N = (L&15)
    const int half = lane >> 4;     // 0 or 1
    const int ln   = lane & 15;
    v8f acc[4][2];
#pragma unroll
    for (int r = 0; r < 4; ++r)
#pragma unroll
        for (int g = 0; g < 2; ++g)
#pragma unroll
            for (int v = 0; v < 8; ++v) acc[r][g][v] = 0.0f;

#pragma unroll
    for (int r = 0; r < 4; ++r) {
#pragma unroll
        for (int t = 0; t < 16; ++t) {
            const int ak = t * 4 + 2 * half;
            v2f a;
            a.x = A[(r * 16 + ln) * 64 + ak];
            a.y = A[(r * 16 + ln) * 64 + ak + 1];
#pragma unroll
            for (int g = 0; g < 2; ++g) {
                v2f bf;
                bf.x = myPsi[ak * 32 + g * 16 + ln];
                bf.y = myPsi[(ak + 1) * 32 + g * 16 + ln];
                acc[r][g] = __builtin_amdgcn_wmma_f32_16x16x4_f32(
                    false, a, false, bf, (short)0, acc[r][g], false, false);
            }
        }
    }

    // ---- ev = psi . Q per column; combine half-wave partial sums -----------
    float ev[2];
#pragma unroll
    for (int g = 0; g < 2; ++g) {
        float s = 0.0f;
#pragma unroll
        for (int r = 0; r < 4; ++r)
#pragma unroll
            for (int v = 0; v < 8; ++v) {
                const int M = r * 16 + v + 8 * half;
                s += acc[r][g][v] * myPsi[M * 32 + g * 16 + ln];
            }
        ev[g] = s;
    }
    float o0 = ev[0] + __shfl_xor(ev[0], 16, 32);
    float o1 = ev[1] + __shfl_xor(ev[1], 16, 32);
    if (lane < 16) {
        const int pbase = blockIdx.x * 128 + wv * 32;
        out[pbase + lane]      = o0;
        out[pbase + 16 + lane] = o1;
    }
}

extern "C" void kernel_launch(void* const* d_in, const int* in_sizes, int n_in,
                              void* d_out, int out_size, void* d_ws, size_t ws_size,
                              hipStream_t stream) {
    const float* inputs  = (const float*)d_in[0];   // (128,32,32,3) f32
    const float* weights = (const float*)d_in[1];   // (24,) f32
    float* out = (float*)d_out;                     // (128,961) f32
    float* A   = (float*)d_ws;                      // 64*64 f32 scratch

    build_A_kernel<<<1, 64, 0, stream>>>(weights, A);
    qev_kernel<<<961, 128, 0, stream>>>(inputs, A, out);
}